// VectorQuantizer_2130303779178
// MI455X (gfx1250) — compile-verified
//
#include <hip/hip_runtime.h>
#include <math.h>

typedef __attribute__((ext_vector_type(2))) float v2f;
typedef __attribute__((ext_vector_type(8))) float v8f;
typedef __attribute__((ext_vector_type(4))) unsigned int u32x4;
typedef __attribute__((ext_vector_type(8))) int i32x8;
typedef __attribute__((ext_vector_type(4))) int i32x4;

#define N_ROWS   16384   // 32*512
#define N_CODES  8192
#define CODE_DIM 256
#define TILE_N   32      // codes per LDS tile (2 WMMA N-tiles)
#define N_TILES  (N_CODES / TILE_N)   // 256
#define ROW_DW   258     // 256 dwords + 2 pad dwords (TDM pad) -> conflict-free b64 reads
#define TILE_DW  (TILE_N * ROW_DW)    // dwords per buffer (8256)

// ---------------- init: zero counts + loss accumulator ----------------
__global__ void k_init(int* __restrict__ counts, float* __restrict__ lossacc) {
    int i = blockIdx.x * blockDim.x + threadIdx.x;
    if (i < N_CODES) counts[i] = 0;
    if (i == N_CODES) *lossacc = 0.0f;
}

// ---------------- per-code squared norms ----------------
__global__ void k_e2(const float* __restrict__ cb, float* __restrict__ e2) {
    int lane = threadIdx.x & 31;
    int wave = threadIdx.x >> 5;
    int code = blockIdx.x * 8 + wave;           // 1024 blocks * 8 waves
    const float4* row = (const float4*)(cb + (size_t)code * CODE_DIM);
    float4 a = row[lane * 2 + 0];
    float4 b = row[lane * 2 + 1];
    float s = a.x*a.x + a.y*a.y + a.z*a.z + a.w*a.w
            + b.x*b.x + b.y*b.y + b.z*b.z + b.w*b.w;
#pragma unroll
    for (int m = 1; m < 32; m <<= 1) s += __shfl_xor(s, m, 32);
    if (lane == 0) e2[code] = s;
}

// ---------------- TDM: async copy of one codebook tile into padded LDS ----------------
// Loads TILE_N rows x 256 dwords from global, padding LDS rows to ROW_DW dwords.
__device__ __forceinline__ void tdm_load_tile(const float* gsrc, unsigned int lds_byte_off) {
    unsigned long long ga = (unsigned long long)(uintptr_t)gsrc;
    u32x4 g0;
    g0[0] = 1u;                                   // count=1 (valid), no gather
    g0[1] = lds_byte_off;                         // lds_addr (bytes)
    g0[2] = (unsigned int)(ga & 0xFFFFFFFFu);     // global_addr[31:0]
    g0[3] = (unsigned int)((ga >> 32) & 0x01FFFFFFu) | (2u << 30);  // addr[56:32] | type=2

    i32x8 g1;
    g1[0] = (2 << 16)        // data_size = 4 bytes
          | (1 << 20)        // pad_enable
          | (7 << 22)        // pad_interval = 256 dwords (one row)
          | (1 << 25);       // pad_amount  = 2 dwords
    g1[1] = (int)((CODE_DIM & 0xFFFF) << 16);     // tensor_dim0[15:0] @ bits[63:48]
    g1[2] = (int)(((unsigned)CODE_DIM >> 16) | ((N_CODES & 0xFFFF) << 16)); // dim0 hi | dim1 lo
    g1[3] = (int)(((unsigned)N_CODES >> 16) | ((unsigned)CODE_DIM << 16));  // dim1 hi | tile_dim0=256
    g1[4] = TILE_N;                               // tile_dim1 = 32, tile_dim2 = 0
    g1[5] = CODE_DIM;                             // tensor_dim0_stride low32 = 256
    g1[6] = 0;                                    // stride0 hi | tensor_dim1_stride lo (unused, 2D tile)
    g1[7] = 0;

    i32x4 gz = {0, 0, 0, 0};
#if defined(__clang_major__) && (__clang_major__ >= 23)
    i32x8 gz8 = {0, 0, 0, 0, 0, 0, 0, 0};
    __builtin_amdgcn_tensor_load_to_lds(g0, g1, gz, gz, gz8, 0);
#else
    __builtin_amdgcn_tensor_load_to_lds(g0, g1, gz, gz, 0);
#endif
}

// ---------------- WMMA distance + argmin ----------------
// grid: 256 blocks of 128 threads (4 waves). Each wave: 16 rows of z.
// Double-buffered TDM tile loads overlap with the WMMA K-loop.
__global__ void __launch_bounds__(128)
k_argmin(const float* __restrict__ z, const float* __restrict__ cb,
         const float* __restrict__ e2, int* __restrict__ idxout) {
    extern __shared__ float lcb[];   // 2 buffers x TILE_N x ROW_DW dwords

    const int t    = threadIdx.x;
    const int lane = t & 31;
    const int wave = t >> 5;
    const int m0   = blockIdx.x * 64 + wave * 16;   // wave's first row
    const int mrow = m0 + (lane & 15);              // A-layout: lane -> M
    const int kofs = 2 * (lane >> 4);               // half-wave -> K pair
    const int ncol = lane & 15;                     // B/C layout: lane -> N

    // A tile (16x256 fp32) resident in VGPRs: areg[i] = z[mrow][4i+kofs .. +1]
    v2f areg[64];
    const float* zr = z + (size_t)mrow * CODE_DIM + kofs;
#pragma unroll
    for (int i = 0; i < 64; ++i)
        areg[i] = *(const v2f*)(zr + 4 * i);

    float bestv[8];
    int   besti[8];
#pragma unroll
    for (int v = 0; v < 8; ++v) { bestv[v] = 3.4e38f; besti[v] = 0; }

    const unsigned int lds_base = (unsigned int)(uintptr_t)&lcb[0];

    // prologue: fetch tile 0 into buffer 0
    if (wave == 0) {
        tdm_load_tile(cb, lds_base);
        __builtin_amdgcn_s_wait_tensorcnt(0);
    }
    __syncthreads();

    for (int nt = 0; nt < N_TILES; ++nt) {
        const int cur = nt & 1;
        const float* bb = lcb + cur * TILE_DW;

        // kick off async TDM copy of the next tile into the other buffer
        if (wave == 0 && nt + 1 < N_TILES)
            tdm_load_tile(cb + (size_t)(nt + 1) * TILE_N * CODE_DIM,
                          lds_base + (unsigned)((cur ^ 1) * TILE_DW * 4));

        v8f acc0 = {};  // S for codes [nt*32 .. +15]
        v8f acc1 = {};  // S for codes [nt*32+16 .. +31]
        const float* b0p = bb + (size_t)ncol * ROW_DW + kofs;
        const float* b1p = bb + (size_t)(ncol + 16) * ROW_DW + kofs;
#pragma unroll
        for (int i = 0; i < 64; ++i) {
            v2f b0 = *(const v2f*)(b0p + 4 * i);   // one ds_load_b64, even-aligned pair
            v2f b1 = *(const v2f*)(b1p + 4 * i);
            acc0 = __builtin_amdgcn_wmma_f32_16x16x4_f32(
                       false, areg[i], false, b0, (short)0, acc0, false, false);
            acc1 = __builtin_amdgcn_wmma_f32_16x16x4_f32(
                       false, areg[i], false, b1, (short)0, acc1, false, false);
        }

        const int   n0 = nt * TILE_N + ncol;
        const int   n1 = n0 + 16;
        const float e0 = e2[n0];
        const float e1 = e2[n1];
#pragma unroll
        for (int v = 0; v < 8; ++v) {
            float d0 = e0 - 2.0f * acc0[v];   // ||z||^2 term is row-constant
            float d1 = e1 - 2.0f * acc1[v];
            if (d0 < bestv[v]) { bestv[v] = d0; besti[v] = n0; }
            if (d1 < bestv[v]) { bestv[v] = d1; besti[v] = n1; }
        }

        // drain the in-flight TDM copy, then publish to all waves
        if (wave == 0 && nt + 1 < N_TILES)
            __builtin_amdgcn_s_wait_tensorcnt(0);
        __syncthreads();
    }

    // argmin reduce across the 16 lanes holding one row; lowest-index tie-break
#pragma unroll
    for (int v = 0; v < 8; ++v) {
        float bv = bestv[v];
        int   bi = besti[v];
#pragma unroll
        for (int m = 1; m < 16; m <<= 1) {
            float ov = __shfl_xor(bv, m, 32);
            int   oi = __shfl_xor(bi, m, 32);
            if (ov < bv || (ov == bv && oi < bi)) { bv = ov; bi = oi; }
        }
        if ((lane & 15) == 0) {
            // C-layout: VGPR v, half-wave h -> row v + 8h
            int r = m0 + v + 8 * (lane >> 4);
            idxout[r] = bi;
        }
    }
}

// ---------------- gather z_q, loss accumulation, histogram ----------------
// grid: 16384 blocks of 64 threads (one row per block, float4 per thread)
__global__ void k_gather(const float* __restrict__ z, const float* __restrict__ cb,
                         const int* __restrict__ idxbuf, float* __restrict__ out,
                         float* __restrict__ lossacc, int* __restrict__ counts) {
    const int row  = blockIdx.x;
    const int t    = threadIdx.x;
    const int code = idxbuf[row];

    const float4* c4 = (const float4*)(cb + (size_t)code * CODE_DIM);
    const float4* z4 = (const float4*)(z  + (size_t)row  * CODE_DIM);
    float4 c  = c4[t];
    float4 zz = z4[t];
    ((float4*)out)[(size_t)row * 64 + t] = c;   // z_q_st == z_q numerically

    float dx = c.x - zz.x, dy = c.y - zz.y, dz = c.z - zz.z, dw = c.w - zz.w;
    float s = dx*dx + dy*dy + dz*dz + dw*dw;
#pragma unroll
    for (int m = 1; m < 32; m <<= 1) s += __shfl_xor(s, m, 32);
    if ((t & 31) == 0) atomicAdd(lossacc, s);

    if (t == 0) {
        atomicAdd(&counts[code], 1);
        out[(size_t)N_ROWS * CODE_DIM + row] = (float)code;  // codes output
    }
}

// ---------------- finalize: loss_vq + perplexity ----------------
__global__ void k_final(const int* __restrict__ counts,
                        const float* __restrict__ lossacc,
                        float* __restrict__ out) {
    __shared__ float red[256];
    int t = threadIdx.x;
    float s = 0.0f;
    for (int i = t; i < N_CODES; i += 256) {
        float avg = (float)counts[i] / (float)N_ROWS;
        s += avg * logf(avg + 1e-10f);
    }
    red[t] = s;
    __syncthreads();
    for (int st = 128; st > 0; st >>= 1) {
        if (t < st) red[t] += red[t + st];
        __syncthreads();
    }
    if (t == 0) {
        const size_t base = (size_t)N_ROWS * CODE_DIM + N_ROWS;
        // loss_codebook == loss_commit numerically -> (1 + beta) * mean
        out[base + 0] = lossacc[0] * 1.25f / (float)((size_t)N_ROWS * CODE_DIM);
        out[base + 1] = expf(-red[0]);
    }
}

extern "C" void kernel_launch(void* const* d_in, const int* in_sizes, int n_in,
                              void* d_out, int out_size, void* d_ws, size_t ws_size,
                              hipStream_t stream) {
    const float* z_e = (const float*)d_in[0];   // (32,512,256) fp32
    const float* cb  = (const float*)d_in[1];   // (8192,256) fp32
    float* out = (float*)d_out;

    // workspace carve-out (all 4-byte aligned)
    char* ws = (char*)d_ws;
    float* e2      = (float*)(ws);                               // 8192 f32
    int*   idxbuf  = (int*)  (ws + 4 * N_CODES);                 // 16384 i32
    int*   counts  = (int*)  (ws + 4 * N_CODES + 4 * N_ROWS);    // 8192 i32
    float* lossacc = (float*)(ws + 4 * N_CODES + 4 * N_ROWS + 4 * N_CODES);

    const size_t lds_bytes = (size_t)2 * TILE_DW * sizeof(float);  // 66048 B

    k_init  <<<(N_CODES + 256) / 256, 256, 0, stream>>>(counts, lossacc);
    k_e2    <<<N_CODES / 8, 256, 0, stream>>>(cb, e2);
    k_argmin<<<N_ROWS / 64, 128, lds_bytes, stream>>>(z_e, cb, e2, idxbuf);
    k_gather<<<N_ROWS, 64, 0, stream>>>(z_e, cb, idxbuf, out, lossacc, counts);
    k_final <<<1, 256, 0, stream>>>(counts, lossacc, out);
}